// SINGLEModel_63385127354436
// MI455X (gfx1250) — compile-verified
//
#include <hip/hip_runtime.h>
#include <hip/hip_bf16.h>

// MI455X (gfx1250): fp32 end-to-end via V_WMMA_F32_16X16X4_F32.
// ~18 GFLOP vs ~200-350 MB HBM traffic @ 23.3 TB/s -> bandwidth-balanced.
// Flash-style online softmax avoids the 524 MB score tensor; K/V and
// encoded_nodes tiles are staged to LDS once per workgroup via the CDNA5
// async global->LDS engine (ASYNCcnt) instead of 16x-duplicated strided loads.

#define B_ 64
#define P_ 256
#define N_ 1000
#define E_ 128
#define H_ 8
#define D_ 16
#define HD_ 128   // H*D

typedef __attribute__((ext_vector_type(2))) float v2f;
typedef __attribute__((ext_vector_type(8))) float v8f;

__device__ __forceinline__ v8f wmma4(v2f a, v2f b, v8f c) {
    // D(16x16) = A(16x4) * B(4x16) + C, fp32
    return __builtin_amdgcn_wmma_f32_16x16x4_f32(
        /*neg_a=*/false, a, /*neg_b=*/false, b,
        /*c_mod=*/(short)0, c, /*reuse_a=*/false, /*reuse_b=*/false);
}

// CDNA5 async global->LDS copies (VGLOBAL GV mode, tracked by ASYNCcnt).
// Low 32 bits of a generic pointer into __shared__ are the LDS byte offset
// (LDS aperture base occupies addr[63:32] only).
__device__ __forceinline__ void async_g2l_b32(void* lds, const void* g) {
    asm volatile("global_load_async_to_lds_b32 %0, %1, off"
                 :: "v"((unsigned)(uintptr_t)lds),
                    "v"((unsigned long long)(uintptr_t)g)
                 : "memory");
}
__device__ __forceinline__ void async_g2l_b128(void* lds, const void* g) {
    asm volatile("global_load_async_to_lds_b128 %0, %1, off"
                 :: "v"((unsigned)(uintptr_t)lds),
                    "v"((unsigned long long)(uintptr_t)g)
                 : "memory");
}
__device__ __forceinline__ void wait_async0() {
    asm volatile("s_wait_asynccnt 0x0" ::: "memory");
}

// ---------------------------------------------------------------------------
// Generic projection GEMM: C[M,128] = A[M,128] @ W[128,128]
//   optional attr rank-1 term (129th input row of W) and optional bias.
// One workgroup (8 waves) per 16-row M tile; wave w owns output cols [16w,16w+16).
// ---------------------------------------------------------------------------
__global__ void proj_gemm(const float* __restrict__ A, const float* __restrict__ W,
                          const float* __restrict__ attr, const float* __restrict__ bias,
                          float* __restrict__ C)
{
    const int lane = threadIdx.x & 31;
    const int wave = threadIdx.x >> 5;      // 0..7
    const int l16  = lane & 15;
    const int half = lane >> 4;
    const int m0   = blockIdx.x * 16;
    const int jcol = wave * 16 + l16;

    const float* arow = A + (size_t)(m0 + l16) * E_ + half * 2;
    const float* wcol = W + (size_t)(half * 2) * E_ + jcol;

    v8f acc = {};
    #pragma unroll
    for (int k0 = 0; k0 < E_; k0 += 4) {
        v2f a; a.x = arow[k0];           a.y = arow[k0 + 1];
        v2f b; b.x = wcol[k0 * E_];      b.y = wcol[k0 * E_ + E_];
        acc = wmma4(a, b, acc);
    }
    if (attr) {  // concat([x, attr]) @ W[129,128]: rank-1 update with W row 128
        const float wlast = W[E_ * E_ + jcol];
        #pragma unroll
        for (int r = 0; r < 8; ++r)
            acc[r] += attr[m0 + half * 8 + r] * wlast;
    }
    if (bias) {
        const float bv = bias[jcol];
        #pragma unroll
        for (int r = 0; r < 8; ++r) acc[r] += bv;
    }
    #pragma unroll
    for (int r = 0; r < 8; ++r)
        C[(size_t)(m0 + half * 8 + r) * E_ + jcol] = acc[r];
}

// ---------------------------------------------------------------------------
// Flash attention. One workgroup (16 waves, 512 thr) = one (b,h); wave w owns
// p-tile [16w,16w+16). Per 16-wide n tile: async-stage K/V tiles to LDS once
// for all 16 waves, 4 wmma for Q.K^T, online softmax, LDS tile transpose of
// the prob tile (wave-private), 4 wmma for P.V.
// ---------------------------------------------------------------------------
__global__ void __launch_bounds__(512)
attn_kernel(const float* __restrict__ q, const float* __restrict__ k,
            const float* __restrict__ v, const float* __restrict__ mask,
            float* __restrict__ out)
{
    __shared__ float kt[16 * 17];            // K tile [n_local][d], padded
    __shared__ float vt[16 * 17];            // V tile [n_local][d], padded
    __shared__ float pt[16 * 16 * 17];       // per-wave prob tiles

    const int tid  = threadIdx.x;
    const int lane = tid & 31;
    const int wave = tid >> 5;               // 0..15 -> p tile
    const int l16  = lane & 15;
    const int half = lane >> 4;
    const int b    = blockIdx.x >> 3;        // grid = B*H
    const int h    = blockIdx.x & 7;
    const int p0   = wave * 16;
    float* wl = pt + wave * (16 * 17);

    const float* qb = q + (size_t)(b * P_ + p0) * HD_ + h * D_;
    const float* kb = k + (size_t)b * N_ * HD_ + h * D_;
    const float* vb = v + (size_t)b * N_ * HD_ + h * D_;
    const float* mb = mask + (size_t)(b * P_ + p0) * N_;

    // staging role: threads [0,256) stage K, [256,512) stage V; 1 float each
    const int  st   = tid & 255;
    const int  sn   = st >> 4;               // tile row 0..15
    const int  sd   = st & 15;               // d column 0..15
    const bool isv  = tid >= 256;
    const float* sb = isv ? vb : kb;
    float* sl       = (isv ? vt : kt) + sn * 17 + sd;

    // Q fragments (A operand: M=p, K=d), reused across all n tiles
    v2f qf[4];
    #pragma unroll
    for (int kk = 0; kk < 4; ++kk) {
        const float* qp = qb + l16 * HD_ + kk * 4 + half * 2;
        qf[kk].x = qp[0]; qf[kk].y = qp[1];
    }

    float m[8], l[8];
    v8f oacc = {};
    #pragma unroll
    for (int r = 0; r < 8; ++r) { m[r] = -3.0e38f; l[r] = 0.0f; }

    for (int n0 = 0; n0 < 1008; n0 += 16) {     // 63 tiles, last is ragged
        // async-stage K/V tiles (ragged rows clamped; clamped values are
        // finite and never consumed: their lanes are forced to -inf / e=0)
        int nrow = n0 + sn; if (nrow > N_ - 1) nrow = N_ - 1;
        async_g2l_b32(sl, sb + (size_t)nrow * HD_ + sd);
        if (n0 + 16 < 1008) {                    // warm L2 for next tile
            int nn = n0 + 16 + sn; if (nn > N_ - 1) nn = N_ - 1;
            __builtin_prefetch(sb + (size_t)nn * HD_ + sd, 0, 0);
        }
        wait_async0();
        __syncthreads();

        const int  n  = n0 + l16;
        const bool nv = (n < N_);

        // scores tile S = Q @ K^T   (B operand: K=d rows, N=n cols)
        v8f s = {};
        #pragma unroll
        for (int kk = 0; kk < 4; ++kk) {
            v2f bf;
            bf.x = kt[l16 * 17 + kk * 4 + half * 2];
            bf.y = kt[l16 * 17 + kk * 4 + half * 2 + 1];
            s = wmma4(qf[kk], bf, s);
        }

        // online softmax (rows half*8+r live in this 16-lane half)
        float e[8];
        #pragma unroll
        for (int r = 0; r < 8; ++r) {
            float val = s[r] * 0.25f;   // 1/sqrt(D)
            val += nv ? mb[(half * 8 + r) * N_ + n] : -3.0e38f;
            float tm = val;
            #pragma unroll
            for (int off = 1; off < 16; off <<= 1)
                tm = fmaxf(tm, __shfl_xor(tm, off, 32));
            const float mn    = fmaxf(m[r], tm);
            const float alpha = __expf(m[r] - mn);
            const float ev    = __expf(val - mn);
            float rs = ev;
            #pragma unroll
            for (int off = 1; off < 16; off <<= 1)
                rs += __shfl_xor(rs, off, 32);
            l[r]    = l[r] * alpha + rs;
            oacc[r] = oacc[r] * alpha;
            m[r]    = mn;
            e[r]    = ev;
        }

        // transpose prob tile through wave-private LDS (C layout -> A layout);
        // DS ops are in-order within a wave, no barrier needed
        #pragma unroll
        for (int r = 0; r < 8; ++r)
            wl[(half * 8 + r) * 17 + l16] = e[r];

        // O += P @ V   (A: M=p K=n from LDS; B: K=n rows, N=d cols)
        #pragma unroll
        for (int kk = 0; kk < 4; ++kk) {
            v2f a;
            a.x = wl[l16 * 17 + kk * 4 + half * 2];
            a.y = wl[l16 * 17 + kk * 4 + half * 2 + 1];
            const int nr0 = kk * 4 + half * 2;
            v2f bf;
            bf.x = vt[(nr0    ) * 17 + l16];
            bf.y = vt[(nr0 + 1) * 17 + l16];
            oacc = wmma4(a, bf, oacc);
        }
        __syncthreads();   // protect kt/vt before next stage overwrites
    }

    #pragma unroll
    for (int r = 0; r < 8; ++r) {
        const int p = p0 + half * 8 + r;
        out[(size_t)(b * P_ + p) * HD_ + h * D_ + l16] = oacc[r] / l[r];
    }
}

// ---------------------------------------------------------------------------
// Final scoring + softmax: probs[b,p,n] = softmax_n(10*tanh(mh.enc/sqrt(E)) + mask)
// One workgroup (16 waves, 512 thr) = one batch b; wave w owns p-tile w.
// The shared 16x128 encoded_nodes tile is async-staged (b128, 16B/thread).
// tanh clips scores to [-10,10] so a fixed max of 10 is numerically exact:
// exp(val-10) in [e^-20, 1]. Streaming pass writes exp, then in-place scale.
// ---------------------------------------------------------------------------
__global__ void __launch_bounds__(512)
final_kernel(const float* __restrict__ mh, const float* __restrict__ enc,
             const float* __restrict__ mask, float* __restrict__ probs)
{
    __shared__ float et[16 * 132];           // enc tile [n_local][e], padded row

    const int tid  = threadIdx.x;
    const int lane = tid & 31;
    const int wave = tid >> 5;               // 0..15 -> p tile
    const int l16  = lane & 15;
    const int half = lane >> 4;
    const int b    = blockIdx.x;             // grid = B
    const int p0   = wave * 16;

    const float* mrow = mh + (size_t)(b * P_ + p0 + l16) * E_ + half * 2;
    const float* eb   = enc + (size_t)b * N_ * E_;
    const float* mb   = mask + (size_t)(b * P_ + p0) * N_;
    float*       ob   = probs + (size_t)(b * P_ + p0) * N_;

    // staging role: thread -> (row tid/32, 4-float group (tid%32)*4), 16B each
    const int sn = tid >> 5;
    const int sc = (tid & 31) * 4;
    float* sl = et + sn * 132 + sc;          // 132*4 and sc*4 are 16B multiples

    // preload all 32 A fragments of the 16x128 mh tile (64 VGPRs)
    v2f af[32];
    #pragma unroll
    for (int kk = 0; kk < 32; ++kk) { af[kk].x = mrow[kk * 4]; af[kk].y = mrow[kk * 4 + 1]; }

    float rs[8];
    #pragma unroll
    for (int r = 0; r < 8; ++r) rs[r] = 0.0f;

    const float inv_sqrt_e = 1.0f / 11.313708498984761f;
    for (int n0 = 0; n0 < 1008; n0 += 16) {
        int nrow = n0 + sn; if (nrow > N_ - 1) nrow = N_ - 1;   // clamp ragged
        async_g2l_b128(sl, eb + (size_t)nrow * E_ + sc);
        if (n0 + 16 < 1008) {
            int nn = n0 + 16 + sn; if (nn > N_ - 1) nn = N_ - 1;
            __builtin_prefetch(eb + (size_t)nn * E_ + sc, 0, 0);
        }
        wait_async0();
        __syncthreads();

        const int  n  = n0 + l16;
        const bool nv = (n < N_);
        v8f s = {};
        #pragma unroll
        for (int kk = 0; kk < 32; ++kk) {
            v2f bf;                          // B operand: K=e rows, N=n cols
            bf.x = et[l16 * 132 + kk * 4 + half * 2];
            bf.y = et[l16 * 132 + kk * 4 + half * 2 + 1];
            s = wmma4(af[kk], bf, s);
        }
        #pragma unroll
        for (int r = 0; r < 8; ++r) {
            float val = 10.0f * tanhf(s[r] * inv_sqrt_e);
            val += nv ? mb[(half * 8 + r) * N_ + n] : -3.0e38f;
            const float ev = __expf(val - 10.0f);
            rs[r] += ev;
            if (nv) ob[(size_t)(half * 8 + r) * N_ + n] = ev;
        }
        __syncthreads();   // protect et before next stage overwrites
    }
    #pragma unroll
    for (int r = 0; r < 8; ++r) {
        #pragma unroll
        for (int off = 1; off < 16; off <<= 1)
            rs[r] += __shfl_xor(rs[r], off, 32);
        rs[r] = 1.0f / rs[r];
    }
    asm volatile("s_wait_storecnt 0x0" ::: "memory");  // pass-1 writes visible
    for (int n0 = 0; n0 < 1008; n0 += 16) {
        const int n = n0 + l16;
        if (n < N_) {
            #pragma unroll
            for (int r = 0; r < 8; ++r) {
                const size_t idx = (size_t)(half * 8 + r) * N_ + n;
                ob[idx] = ob[idx] * rs[r];
            }
        }
    }
}

extern "C" void kernel_launch(void* const* d_in, const int* in_sizes, int n_in,
                              void* d_out, int out_size, void* d_ws, size_t ws_size,
                              hipStream_t stream) {
    (void)in_sizes; (void)n_in; (void)out_size; (void)ws_size;
    const float* enc  = (const float*)d_in[0];  // [B,N,E]
    const float* eln  = (const float*)d_in[1];  // [B,P,E]
    const float* attr = (const float*)d_in[2];  // [B,P,1]
    const float* mask = (const float*)d_in[3];  // [B,P,N]
    const float* Wq   = (const float*)d_in[4];  // [E+1, H*D]
    const float* Wk   = (const float*)d_in[5];  // [E, H*D]
    const float* Wv   = (const float*)d_in[6];  // [E, H*D]
    const float* Wo   = (const float*)d_in[7];  // [H*D, E]
    const float* bo   = (const float*)d_in[8];  // [E]
    float* out = (float*)d_out;                 // [B,P,N]

    float* ws   = (float*)d_ws;                 // ~91 MB fp32 scratch
    float* kbuf = ws;                           // [B*N,128]
    float* vbuf = kbuf + (size_t)B_ * N_ * E_;  // [B*N,128]
    float* qbuf = vbuf + (size_t)B_ * N_ * E_;  // [B*P,128]
    float* abuf = qbuf + (size_t)B_ * P_ * E_;  // [B*P,128] attention out
    float* mbuf = abuf + (size_t)B_ * P_ * E_;  // [B*P,128] mh

    proj_gemm<<<(B_ * N_) / 16, 256, 0, stream>>>(enc, Wk, nullptr, nullptr, kbuf);
    proj_gemm<<<(B_ * N_) / 16, 256, 0, stream>>>(enc, Wv, nullptr, nullptr, vbuf);
    proj_gemm<<<(B_ * P_) / 16, 256, 0, stream>>>(eln, Wq, attr, nullptr, qbuf);
    attn_kernel<<<B_ * H_, 512, 0, stream>>>(qbuf, kbuf, vbuf, mask, abuf);
    proj_gemm<<<(B_ * P_) / 16, 256, 0, stream>>>(abuf, Wo, nullptr, bo, mbuf);
    final_kernel<<<B_, 512, 0, stream>>>(mbuf, enc, mask, out);
}